// AttZipLlamaAttention_streaming_37666863186509
// MI455X (gfx1250) — compile-verified
//
#include <hip/hip_runtime.h>
#include <hip/hip_bf16.h>

// ---------------------------------------------------------------------------
// AttZip LLaMA decode-step attention + importance eviction for gfx1250.
// B=8 H=32 S=4096 hd=128 D=4096, q_len=1, kv_len=4097, keep 4 topk + 512 recent.
// ---------------------------------------------------------------------------

#define B_      8
#define H_      32
#define S_      4096
#define HD_     128
#define D_      4096
#define KV_     4097
#define RECENT_ 512
#define CAND_   (KV_ - RECENT_)   // 3585
#define KEEP_   516

typedef float v2f __attribute__((ext_vector_type(2)));
typedef float v8f __attribute__((ext_vector_type(8)));

// ---------------------------------------------------------------------------
// Padded-M GEMM via V_WMMA_F32_16X16X4_F32 (exact fp32).
//   out[0..7][N] = A[0..7][K] * W[K][N],  A padded to 16 rows (rows 8..15 zero)
// One wave owns a 16-wide column tile, loops K=4096 in steps of 4.
// A frag (16x4 f32): lane l<16 -> (M=l, K=0/1), lane 16+l -> (M=l, K=2/3)
// B frag (4x16 f32): lane l<16 -> (K=0/1, N=l), lane 16+l -> (K=2/3, N=l)
// D (16x16 f32):     lanes 0-15 hold M=0..7 in VGPRs 0..7 -> our 8 batch rows.
// ---------------------------------------------------------------------------
__global__ __launch_bounds__(256) void gemm16_wmma(const float* __restrict__ A,
                                                   const float* __restrict__ W,
                                                   float* __restrict__ out)
{
    const int lane = threadIdx.x & 31;
    const int wave = threadIdx.x >> 5;
    const int cb   = (blockIdx.x * 8 + wave) * 16;   // column tile base, < 4096
    const int m    = lane & 15;
    const int kr   = (lane >> 4) << 1;               // 0 or 2

    const float* pa  = A + m * D_ + kr;              // A[m][kb+kr], A[m][kb+kr+1]
    const float* pb0 = W + kr * D_ + cb + m;         // W[kb+kr  ][cb+n], n==m
    const float* pb1 = pb0 + D_;                     // W[kb+kr+1][cb+n]

    v8f c = {0.f, 0.f, 0.f, 0.f, 0.f, 0.f, 0.f, 0.f};
#pragma unroll 4
    for (int kb = 0; kb < D_; kb += 4) {
        v2f a; a.x = pa[0];  a.y = pa[1];
        v2f b; b.x = pb0[0]; b.y = pb1[0];
        c = __builtin_amdgcn_wmma_f32_16x16x4_f32(false, a, false, b,
                                                  (short)0, c, false, false);
        pa  += 4;
        pb0 += 4 * D_;
        pb1 += 4 * D_;
    }
    if (lane < 16) {
#pragma unroll
        for (int r = 0; r < 8; ++r)                  // rows 8..15 are padding
            out[r * D_ + cb + lane] = c[r];
    }
}

// ---------------------------------------------------------------------------
// Stage hidden into a 16-row padded buffer; zero pad rows of the ctx buffer.
// grid 256 x 256 threads covers 16*4096 elements.
// ---------------------------------------------------------------------------
__global__ __launch_bounds__(256) void init_pads(const float* __restrict__ hidden,
                                                 float* __restrict__ hpad,
                                                 float* __restrict__ cpad)
{
    const int i   = blockIdx.x * 256 + threadIdx.x;
    const int row = i >> 12;                         // /4096
    hpad[i] = (row < B_) ? hidden[i] : 0.f;
    if (row >= B_) cpad[i] = 0.f;
}

// ---------------------------------------------------------------------------
// RoPE for the new token at position S. Layout of q/k: [(b*H+h)*128 + d].
// One block per (b,h); thread i handles the pair (d=i, d=i+64).
// ---------------------------------------------------------------------------
__global__ __launch_bounds__(64) void rope_kernel(float* __restrict__ q,
                                                  float* __restrict__ k)
{
    const int bh   = blockIdx.x;
    const int i    = threadIdx.x;                    // 0..63
    const int base = bh * HD_;

    const float inv_freq = __expf(-(float)i * (logf(10000.0f) / 64.0f));
    const float ang = (float)S_ * inv_freq;
    const float c = cosf(ang), s = sinf(ang);

    float q0 = q[base + i], q1 = q[base + i + 64];
    q[base + i]      = q0 * c - q1 * s;
    q[base + i + 64] = q1 * c + q0 * s;
    float k0 = k[base + i], k1 = k[base + i + 64];
    k[base + i]      = k0 * c - k1 * s;
    k[base + i + 64] = k1 * c + k0 * s;
}

// ---------------------------------------------------------------------------
// Decode attention: one workgroup per (b,h), 16 waves x 256 keys.
// Pass 1: scores (coalesced float4 row loads + shfl_xor reduction) -> LDS, max.
// Pass 2: p = exp(s - M) (stored to pbuf for importance), l-sum, ctx += p*V.
// Final: cross-wave ctx reduction, ctx/L -> padded ctx buffer, L -> Lbuf.
// ---------------------------------------------------------------------------
__global__ __launch_bounds__(512) void attn_decode(const float* __restrict__ qv,
                                                   const float* __restrict__ knew,
                                                   const float* __restrict__ vnew,
                                                   const float* __restrict__ past_k,
                                                   const float* __restrict__ past_v,
                                                   float* __restrict__ pbuf,
                                                   float* __restrict__ Lbuf,
                                                   float* __restrict__ cpad)
{
    __shared__ float sc[KV_];
    __shared__ float wred[16];
    __shared__ float cred[16 * HD_];
    __shared__ float Msh, Lsh;

    const int bh   = blockIdx.x;                     // 0..255
    const int b    = bh >> 5;
    const int h    = bh & 31;
    const int lane = threadIdx.x & 31;
    const int wave = threadIdx.x >> 5;

    const float scale = 0.08838834764831845f;        // 1/sqrt(128)
    const float* qp = qv + bh * HD_ + lane * 4;
    const float q0 = qp[0] * scale, q1 = qp[1] * scale;
    const float q2 = qp[2] * scale, q3 = qp[3] * scale;

    const float* Kbase = past_k + (size_t)bh * S_ * HD_;
    const float* Vbase = past_v + (size_t)bh * S_ * HD_;
    const int k0 = wave * 256;

    // ---- pass 1: scores + max ----
    float lmax = -3.4e38f;
    for (int kk = 0; kk < 256; ++kk) {
        const int key = k0 + kk;
        const float4 kr = *(const float4*)(Kbase + (size_t)key * HD_ + lane * 4);
        float p = q0 * kr.x + q1 * kr.y + q2 * kr.z + q3 * kr.w;
        for (int off = 16; off; off >>= 1) p += __shfl_xor(p, off, 32);
        if (lane == 0) sc[key] = p;
        lmax = fmaxf(lmax, p);
    }
    if (wave == 0) {                                 // new token key (index 4096)
        const float4 kr = *(const float4*)(knew + bh * HD_ + lane * 4);
        float p = q0 * kr.x + q1 * kr.y + q2 * kr.z + q3 * kr.w;
        for (int off = 16; off; off >>= 1) p += __shfl_xor(p, off, 32);
        if (lane == 0) sc[S_] = p;
        lmax = fmaxf(lmax, p);
    }
    if (lane == 0) wred[wave] = lmax;
    __syncthreads();
    if (threadIdx.x == 0) {
        float m = wred[0];
        for (int w = 1; w < 16; ++w) m = fmaxf(m, wred[w]);
        Msh = m;
    }
    __syncthreads();
    const float M = Msh;

    // ---- pass 2: exp weights, l-sum, ctx ----
    float4 ctx = {0.f, 0.f, 0.f, 0.f};
    float lsum = 0.f;
    float* pout = pbuf + (size_t)bh * KV_;
    for (int kk = 0; kk < 256; ++kk) {
        const int key = k0 + kk;
        const float p = expf(sc[key] - M);
        if (lane == 0) pout[key] = p;
        lsum += p;
        const float4 vr = *(const float4*)(Vbase + (size_t)key * HD_ + lane * 4);
        ctx.x += p * vr.x; ctx.y += p * vr.y;
        ctx.z += p * vr.z; ctx.w += p * vr.w;
    }
    if (wave == 0) {
        const float p = expf(sc[S_] - M);
        if (lane == 0) pout[S_] = p;
        lsum += p;
        const float4 vr = *(const float4*)(vnew + bh * HD_ + lane * 4);
        ctx.x += p * vr.x; ctx.y += p * vr.y;
        ctx.z += p * vr.z; ctx.w += p * vr.w;
    }
    if (lane == 0) wred[wave] = lsum;
    cred[wave * HD_ + lane * 4 + 0] = ctx.x;
    cred[wave * HD_ + lane * 4 + 1] = ctx.y;
    cred[wave * HD_ + lane * 4 + 2] = ctx.z;
    cred[wave * HD_ + lane * 4 + 3] = ctx.w;
    __syncthreads();
    if (threadIdx.x == 0) {
        float L = 0.f;
        for (int w = 0; w < 16; ++w) L += wred[w];
        Lsh = L;
        Lbuf[bh] = L;
    }
    __syncthreads();
    if (threadIdx.x < HD_) {
        float acc = 0.f;
        for (int w = 0; w < 16; ++w) acc += cred[w * HD_ + threadIdx.x];
        cpad[b * D_ + h * HD_ + threadIdx.x] = acc / Lsh;
    }
}

// ---------------------------------------------------------------------------
// Importance scores, top-4 (lowest-index tie-break like lax.top_k), keep list,
// imp_kept output. One block per batch b.
// ---------------------------------------------------------------------------
__global__ __launch_bounds__(256) void importance_topk(const float* __restrict__ pbuf,
                                                       const float* __restrict__ Lbuf,
                                                       const float* __restrict__ prev,
                                                       int* __restrict__ keep,
                                                       float* __restrict__ imp_kept)
{
    __shared__ float simp[KV_];
    __shared__ float invL[H_];
    __shared__ float bv[256];
    __shared__ int   bi[256];
    __shared__ int   sel[4];

    const int b = blockIdx.x;
    const int t = threadIdx.x;

    if (t < H_) invL[t] = 1.0f / ((float)H_ * Lbuf[b * H_ + t]);
    __syncthreads();

    for (int k = t; k < KV_; k += 256) {
        float s = 0.f;
        for (int h = 0; h < H_; ++h)
            s += pbuf[(size_t)(b * H_ + h) * KV_ + k] * invL[h];
        if (k < S_) s += prev[b * S_ + k];
        simp[k] = s;
    }
    __syncthreads();

    for (int it = 0; it < 4; ++it) {
        float bestv = -3.4e38f;
        int   besti = 0x7fffffff;
        for (int k = t; k < CAND_; k += 256) {
            bool skip = false;
            for (int j = 0; j < it; ++j) if (sel[j] == k) skip = true;
            if (skip) continue;
            const float v = simp[k];
            if (v > bestv || (v == bestv && k < besti)) { bestv = v; besti = k; }
        }
        bv[t] = bestv; bi[t] = besti;
        __syncthreads();
        for (int s2 = 128; s2; s2 >>= 1) {
            if (t < s2) {
                const float v2 = bv[t + s2];
                const int   i2 = bi[t + s2];
                if (v2 > bv[t] || (v2 == bv[t] && i2 < bi[t])) { bv[t] = v2; bi[t] = i2; }
            }
            __syncthreads();
        }
        if (t == 0) sel[it] = bi[0];
        __syncthreads();
    }
    if (t == 0) {                                    // sort 4 indices ascending
        for (int a = 0; a < 3; ++a)
            for (int c2 = 0; c2 < 3 - a; ++c2)
                if (sel[c2] > sel[c2 + 1]) { int tmp = sel[c2]; sel[c2] = sel[c2 + 1]; sel[c2 + 1] = tmp; }
    }
    __syncthreads();
    for (int j = t; j < KEEP_; j += 256) {
        const int src = (j < 4) ? sel[j] : (CAND_ - 4 + j); // j>=4 -> 3585..4096
        keep[b * KEEP_ + j]     = src;
        imp_kept[b * KEEP_ + j] = simp[src];
    }
}

// ---------------------------------------------------------------------------
// Gather kept K/V rows. One wave per output row (b,h,j); float4 copies.
// grid = 8*32*516/8 = 16512 blocks of 256 threads (8 waves).
// ---------------------------------------------------------------------------
__global__ __launch_bounds__(256) void gather_kv(const float* __restrict__ past_k,
                                                 const float* __restrict__ past_v,
                                                 const float* __restrict__ knew,
                                                 const float* __restrict__ vnew,
                                                 const int* __restrict__ keep,
                                                 float* __restrict__ kout,
                                                 float* __restrict__ vout)
{
    const int row  = blockIdx.x * 8 + (threadIdx.x >> 5);
    const int lane = threadIdx.x & 31;
    const int j  = row % KEEP_;
    const int bh = row / KEEP_;
    const int b  = bh >> 5;
    const int src = keep[b * KEEP_ + j];

    const float4* ks;
    const float4* vs;
    if (src < S_) {
        ks = (const float4*)(past_k + ((size_t)bh * S_ + src) * HD_);
        vs = (const float4*)(past_v + ((size_t)bh * S_ + src) * HD_);
    } else {
        ks = (const float4*)(knew + bh * HD_);
        vs = (const float4*)(vnew + bh * HD_);
    }
    float4* kd = (float4*)(kout + ((size_t)bh * KEEP_ + j) * HD_);
    float4* vd = (float4*)(vout + ((size_t)bh * KEEP_ + j) * HD_);
    kd[lane] = ks[lane];
    vd[lane] = vs[lane];
}

// ---------------------------------------------------------------------------
extern "C" void kernel_launch(void* const* d_in, const int* in_sizes, int n_in,
                              void* d_out, int out_size, void* d_ws, size_t ws_size,
                              hipStream_t stream)
{
    (void)in_sizes; (void)n_in; (void)out_size; (void)ws_size;

    const float* hidden = (const float*)d_in[0];
    const float* past_k = (const float*)d_in[1];
    const float* past_v = (const float*)d_in[2];
    const float* Wq     = (const float*)d_in[3];
    const float* Wk     = (const float*)d_in[4];
    const float* Wv     = (const float*)d_in[5];
    const float* Wo     = (const float*)d_in[6];
    const float* prev   = (const float*)d_in[7];

    // outputs, flat fp32: attn_output | k_ev | v_ev | imp_kept
    float* out      = (float*)d_out;
    float* attn_out = out;                               // 8*4096
    float* k_ev     = attn_out + (B_ * D_);              // 8*32*516*128
    float* v_ev     = k_ev + (B_ * H_ * KEEP_ * HD_);
    float* imp_kept = v_ev + (B_ * H_ * KEEP_ * HD_);    // 8*516

    // workspace layout (~5.1 MB)
    float* ws   = (float*)d_ws;
    float* qbuf = ws;                                    // B*D
    float* kbuf = qbuf + B_ * D_;                        // B*D
    float* vbuf = kbuf + B_ * D_;                        // B*D
    float* hpad = vbuf + B_ * D_;                        // 16*D
    float* cpad = hpad + 16 * D_;                        // 16*D
    float* pbuf = cpad + 16 * D_;                        // B*H*KV
    float* Lbuf = pbuf + (size_t)B_ * H_ * KV_;          // B*H
    int*   keep = (int*)(Lbuf + B_ * H_);                // B*KEEP

    init_pads<<<256, 256, 0, stream>>>(hidden, hpad, cpad);
    gemm16_wmma<<<32, 256, 0, stream>>>(hpad, Wq, qbuf);
    gemm16_wmma<<<32, 256, 0, stream>>>(hpad, Wk, kbuf);
    gemm16_wmma<<<32, 256, 0, stream>>>(hpad, Wv, vbuf);
    rope_kernel<<<B_ * H_, 64, 0, stream>>>(qbuf, kbuf);
    attn_decode<<<B_ * H_, 512, 0, stream>>>(qbuf, kbuf, vbuf, past_k, past_v,
                                             pbuf, Lbuf, cpad);
    gemm16_wmma<<<32, 256, 0, stream>>>(cpad, Wo, attn_out);
    importance_topk<<<B_, 256, 0, stream>>>(pbuf, Lbuf, prev, keep, imp_kept);
    gather_kv<<<(B_ * H_ * KEEP_) / 8, 256, 0, stream>>>(past_k, past_v, kbuf, vbuf,
                                                         keep, k_ev, v_ev);
}